// BasicDeconvolutionBlock_7378753815236
// MI455X (gfx1250) — compile-verified
//
#include <hip/hip_runtime.h>
#include <hip/hip_bf16.h>

typedef float v2f __attribute__((ext_vector_type(2)));
typedef float v8f __attribute__((ext_vector_type(8)));

#define KOFF 27
#define CIN 64
#define COUT 64
#define LDSW 80   // LDS row stride in dwords: 2-row delta = 160 % 64 banks = 32 -> conflict-free half-wave split
#define BN_EPS 1e-5f

// ---------------------------------------------------------------------------
// Kernel 0: zero the accumulator (d_out) and the BN stats buffer (d_ws).
// ---------------------------------------------------------------------------
__global__ __launch_bounds__(256) void zero_kernel(float* __restrict__ out, int n4,
                                                   float* __restrict__ stats) {
    float4 z = make_float4(0.f, 0.f, 0.f, 0.f);
    for (int i = blockIdx.x * blockDim.x + threadIdx.x; i < n4; i += gridDim.x * blockDim.x)
        reinterpret_cast<float4*>(out)[i] = z;
    if (blockIdx.x == 0 && threadIdx.x < 2 * COUT) stats[threadIdx.x] = 0.f;
}

// ---------------------------------------------------------------------------
// Kernel 1: gather -> WMMA f32 GEMM -> scatter-add (atomic f32).
// blockIdx.y = kernel offset k (0..26); blockIdx.x covers M in 128-row tiles.
// Each wave32 computes a 16x64 output tile via V_WMMA_F32_16X16X4_F32.
// ---------------------------------------------------------------------------
__global__ __launch_bounds__(256)
void spconv_wmma_kernel(const float* __restrict__ feats,
                        const float* __restrict__ W,
                        const int* __restrict__ in_map,
                        const int* __restrict__ out_map,
                        float* __restrict__ out,
                        int M) {
    __shared__ float Wlds[CIN * LDSW];

    const int k = blockIdx.y;
    const int t = threadIdx.x;

    // Stage W[k] (64x64 f32, row-major [cin][cout]) into padded LDS.
    const float4* Wg = reinterpret_cast<const float4*>(W + (size_t)k * CIN * COUT);
#pragma unroll
    for (int it = 0; it < 4; ++it) {
        int flat = it * 1024 + t * 4;            // element index within 64x64
        float4 w4 = Wg[flat >> 2];
        int c = flat >> 6, d = flat & 63;
        float* dst = &Wlds[c * LDSW + d];
        dst[0] = w4.x; dst[1] = w4.y; dst[2] = w4.z; dst[3] = w4.w;
    }
    __syncthreads();

    const int lane = t & 31;
    const int hi   = lane >> 4;     // half-wave select: 0 -> K{0,1}, 1 -> K{2,3}
    const int l16  = lane & 15;
    const int wave = t >> 5;
    const int m_base = blockIdx.x * 128 + wave * 16;

    // A layout (ISA 16x4 f32): lanes 0-15 and 16-31 both hold row M=lane%16,
    // VGPR0/1 = K0/K1 (low half) or K2/K3 (high half).
    const int m_row = m_base + l16;
    const int mc    = (m_row < M) ? m_row : (M - 1);     // clamp: EXEC stays all-1s
    const float* arow = feats + (size_t)in_map[(size_t)k * M + mc] * CIN;

    v2f a[16];
#pragma unroll
    for (int s = 0; s < 16; ++s) {
        float4 f4 = reinterpret_cast<const float4*>(arow)[s];   // cin 4s..4s+3
        a[s].x = hi ? f4.z : f4.x;
        a[s].y = hi ? f4.w : f4.y;
    }

    v8f acc[4] = {v8f{}, v8f{}, v8f{}, v8f{}};
#pragma unroll
    for (int s = 0; s < 16; ++s) {
        // B layout: VGPR0 = row K(2*hi), VGPR1 = row K(2*hi+1), lane%16 = col.
        const float* b0 = &Wlds[(4 * s + 2 * hi) * LDSW];
        const float* b1 = b0 + LDSW;
#pragma unroll
        for (int nb = 0; nb < 4; ++nb) {
            v2f b;
            b.x = b0[nb * 16 + l16];
            b.y = b1[nb * 16 + l16];
            acc[nb] = __builtin_amdgcn_wmma_f32_16x16x4_f32(
                false, a[s], false, b, (short)0, acc[nb], false, false);
        }
    }

    // Scatter-add. C/D layout: VGPR v holds row (v + 8*hi), col = l16 (+16*nb).
    long drow[8];
#pragma unroll
    for (int v = 0; v < 8; ++v) {
        int m = m_base + v + 8 * hi;
        drow[v] = (m < M) ? (long)out_map[(size_t)k * M + m] * COUT : -1L;
    }
#pragma unroll
    for (int nb = 0; nb < 4; ++nb) {
#pragma unroll
        for (int v = 0; v < 8; ++v) {
            if (drow[v] >= 0)
                atomicAdd(&out[drow[v] + nb * 16 + l16], acc[nb][v]);
        }
    }
}

// ---------------------------------------------------------------------------
// Kernel 2: per-channel sum / sum-of-squares over out[n_out][64].
// ---------------------------------------------------------------------------
__global__ __launch_bounds__(256)
void stats_kernel(const float* __restrict__ out, float* __restrict__ stats, int n_out) {
    __shared__ float s_sum[256];
    __shared__ float s_sq[256];
    const int t = threadIdx.x;
    const int c = t & 63;
    const int g = t >> 6;   // 0..3: four rows in flight per block
    float sum = 0.f, sq = 0.f;
    const int rend = min(n_out, (int)(blockIdx.x + 1) * 256);
    for (int r = blockIdx.x * 256 + g; r < rend; r += 4) {
        float v = out[(size_t)r * COUT + c];
        sum += v;
        sq  += v * v;
    }
    s_sum[t] = sum;
    s_sq[t]  = sq;
    __syncthreads();
    if (t < 64) {
        float ts = s_sum[t] + s_sum[64 + t] + s_sum[128 + t] + s_sum[192 + t];
        float tq = s_sq[t]  + s_sq[64 + t]  + s_sq[128 + t]  + s_sq[192 + t];
        atomicAdd(&stats[t], ts);
        atomicAdd(&stats[64 + t], tq);
    }
}

// ---------------------------------------------------------------------------
// Kernel 3: in-place BatchNorm (population stats) + ReLU.
// ---------------------------------------------------------------------------
__global__ __launch_bounds__(256)
void bn_relu_kernel(float* __restrict__ out, const float* __restrict__ stats,
                    const float* __restrict__ gamma, const float* __restrict__ beta,
                    int n_out) {
    const float inv_n = 1.0f / (float)n_out;
    const long total4 = (long)n_out * (COUT / 4);
    for (long i = blockIdx.x * blockDim.x + threadIdx.x; i < total4;
         i += (long)gridDim.x * blockDim.x) {
        float4 x = reinterpret_cast<float4*>(out)[i];
        const int c0 = (int)((i * 4) & 63);
        float r[4] = {x.x, x.y, x.z, x.w};
#pragma unroll
        for (int j = 0; j < 4; ++j) {
            const int c = c0 + j;
            float mean = stats[c] * inv_n;
            float var  = stats[64 + c] * inv_n - mean * mean;
            float y = (r[j] - mean) * rsqrtf(var + BN_EPS) * gamma[c] + beta[c];
            r[j] = fmaxf(y, 0.f);
        }
        x.x = r[0]; x.y = r[1]; x.z = r[2]; x.w = r[3];
        reinterpret_cast<float4*>(out)[i] = x;
    }
}

// ---------------------------------------------------------------------------
extern "C" void kernel_launch(void* const* d_in, const int* in_sizes, int n_in,
                              void* d_out, int out_size, void* d_ws, size_t ws_size,
                              hipStream_t stream) {
    const float* feats   = (const float*)d_in[0];
    const float* W       = (const float*)d_in[1];
    const float* gamma   = (const float*)d_in[2];
    const float* beta    = (const float*)d_in[3];
    const int*   in_map  = (const int*)d_in[4];
    const int*   out_map = (const int*)d_in[5];
    float* out   = (float*)d_out;
    float* stats = (float*)d_ws;          // 128 floats: [sum(64), sumsq(64)]

    const int M     = in_sizes[4] / KOFF; // 50000
    const int n_out = out_size / COUT;    // 200000

    // 1) zero accumulator + stats
    zero_kernel<<<1024, 256, 0, stream>>>(out, out_size / 4, stats);

    // 2) gather-GEMM-scatter with f32 WMMA
    dim3 grid((M + 127) / 128, KOFF);
    spconv_wmma_kernel<<<grid, 256, 0, stream>>>(feats, W, in_map, out_map, out, M);

    // 3) per-channel stats
    int sblocks = (n_out + 255) / 256;
    stats_kernel<<<sblocks, 256, 0, stream>>>(out, stats, n_out);

    // 4) BN + ReLU in place
    bn_relu_kernel<<<4096, 256, 0, stream>>>(out, stats, gamma, beta, n_out);
}